// LocallyDirected1D_67585605370330
// MI455X (gfx1250) — compile-verified
//
#include <hip/hip_runtime.h>

// ---------------------------------------------------------------------------
// LocallyDirected1D  (B=64, N_IN=1e6, N_OUT=20000, CIN=FILTERS=1)
//
//   out[b, s] = bias[s] + sum_{e : mask_col[e]==s} x[b, mask_row[e]] * kernel[e]
//
// Roofline (MI455X): 128 MFLOP vs ~273 MB compulsory HBM traffic (x = 256 MB
// read exactly once; kernel+masks = 12 MB, L2-resident in the 192 MB L2 across
// the 64 batch re-reads; out = 5 MB). At 23.3 TB/s the floor is ~12 us ->
// purely bandwidth bound; WMMA is inapplicable at 0.5 FLOP/byte.
//
// Instruction accounting (round-1 lesson): one wave32 per (b,s) pays ~40
// fixed instructions per 50-element segment -> ~54M wave-instr, issue-bound.
// This version uses ONE THREAD per (b,s): lanes cover 32 consecutive segments
// of the same batch (n_seg % 32 == 0 -> b is wave-uniform), each lane streams
// its contiguous e-range with b128 loads of mask_row/kernel and a runtime-
// verified float4 fast path for the x gather (mask_row is an arange in
// practice; generic scalar fallback otherwise). ~13M wave-instr total ->
// under the HBM floor. No atomics, fixed per-lane summation order ->
// bitwise deterministic.
//
// mask_col is SORTED, so each segment is a contiguous e-range:
//   Pass 1: seg_start[s] = lower_bound(mask_col, s), s in [0, n_seg]  (d_ws)
//   Pass 2: per-(b,s) vectorized dot-product over [seg_start[s], seg_start[s+1])
// ---------------------------------------------------------------------------

__global__ void seg_bounds_kernel(const int* __restrict__ mask_col,
                                  int* __restrict__ seg_start,
                                  int n_elem, int n_seg) {
    int s = blockIdx.x * blockDim.x + threadIdx.x;
    if (s > n_seg) return;
    int lo = 0, hi = n_elem;
    while (lo < hi) {
        int mid = (lo + hi) >> 1;
        if (mask_col[mid] < s) lo = mid + 1; else hi = mid;
    }
    seg_start[s] = lo;   // lower bound of segment s; seg_start[n_seg] == n_elem
}

__global__ void __launch_bounds__(256)
seg_dot_kernel(const float* __restrict__ x,
               const float* __restrict__ kern,
               const float* __restrict__ bias,
               const int* __restrict__ mask_row,
               const int* __restrict__ seg_start,
               float* __restrict__ out,
               int n_elem, int n_seg, int n_batch) {
    const int tid = blockIdx.x * blockDim.x + threadIdx.x;   // < 1.28M, fits int
    if (tid >= n_batch * n_seg) return;
    const int b = tid / n_seg;            // wave-uniform (n_seg % 32 == 0)
    const int s = tid - b * n_seg;        // lanes -> 32 consecutive segments

    int e        = seg_start[s];
    const int e1 = seg_start[s + 1];
    const float* xb = x + (size_t)b * (size_t)n_elem;
    float acc = 0.0f;

    // Head: align e to 4 so the b128 loads below are 16B-aligned.
    for (; e < e1 && (e & 3); ++e)
        acc = __builtin_fmaf(xb[mask_row[e]], kern[e], acc);

    // Main: 4 elements per step. mask_row/kern are contiguous per lane ->
    // one global_load_b128 each. x gather collapses to one b128 when the 4
    // rows are consecutive and 16B-aligned (always true for arange mask_row).
    for (; e + 4 <= e1; e += 4) {
        const int4   r = *reinterpret_cast<const int4*>(mask_row + e);
        const float4 k = *reinterpret_cast<const float4*>(kern + e);

        if ((e & 15) == 0 && r.x + 259 < n_elem)
            __builtin_prefetch(xb + r.x + 256, 0, 0);   // gfx1250 global_prefetch_b8

        if (((r.x & 3) == 0) & (r.y == r.x + 1) & (r.z == r.x + 2) & (r.w == r.x + 3)) {
            const float4 xv = *reinterpret_cast<const float4*>(xb + r.x);
            acc = __builtin_fmaf(xv.x, k.x, acc);
            acc = __builtin_fmaf(xv.y, k.y, acc);
            acc = __builtin_fmaf(xv.z, k.z, acc);
            acc = __builtin_fmaf(xv.w, k.w, acc);
        } else {
            acc = __builtin_fmaf(xb[r.x], k.x, acc);
            acc = __builtin_fmaf(xb[r.y], k.y, acc);
            acc = __builtin_fmaf(xb[r.z], k.z, acc);
            acc = __builtin_fmaf(xb[r.w], k.w, acc);
        }
    }

    // Tail
    for (; e < e1; ++e)
        acc = __builtin_fmaf(xb[mask_row[e]], kern[e], acc);

    out[(size_t)b * (size_t)n_seg + s] = acc + bias[s];   // coalesced across lanes
}

// ---------------------------------------------------------------------------
// Never launched. Compile-only probe for gfx1250-specific codegen paths:
// cluster-id TTMP reads, s_barrier_signal/wait -3, s_wait_tensorcnt, and the
// async-to-LDS data mover + s_wait_asynccnt. Round-2 diagnostic showed the
// async builtin exists with param 1 = AS1 pointer to a 16B int vector.
// ---------------------------------------------------------------------------
typedef int cdna5_v4i __attribute__((vector_size(4 * sizeof(int))));

__global__ void cdna5_isa_probe(int* p, const int* g) {
    int cid = __builtin_amdgcn_cluster_id_x();
    __builtin_amdgcn_s_cluster_barrier();
    __builtin_amdgcn_s_wait_tensorcnt(0);
#if __has_builtin(__builtin_amdgcn_global_load_async_to_lds_b128)
    __shared__ cdna5_v4i stage[16];
    __builtin_amdgcn_global_load_async_to_lds_b128(
        (__attribute__((address_space(1))) cdna5_v4i*)g,
        (__attribute__((address_space(3))) cdna5_v4i*)stage,
        /*offset=*/0, /*cpol=*/0);
#if __has_builtin(__builtin_amdgcn_s_wait_asynccnt)
    __builtin_amdgcn_s_wait_asynccnt(0);
#endif
    __syncthreads();
    cid += ((const int*)stage)[threadIdx.x & 63u];
#endif
    p[threadIdx.x] = cid;
}

extern "C" void kernel_launch(void* const* d_in, const int* in_sizes, int n_in,
                              void* d_out, int out_size, void* d_ws, size_t ws_size,
                              hipStream_t stream) {
    const float* x        = (const float*)d_in[0];  // (B, N_IN, 1) f32
    const float* kern     = (const float*)d_in[1];  // (N_IN, 1)   f32
    const float* bias     = (const float*)d_in[2];  // (N_OUT, 1)  f32
    const int*   mask_row = (const int*)d_in[3];    // (N_IN,) i32
    const int*   mask_col = (const int*)d_in[4];    // (N_IN,) i32 sorted
    float* out = (float*)d_out;                     // (B, N_OUT, 1) f32

    const int n_elem  = in_sizes[1];                // N_IN  = 1,000,000
    const int n_seg   = in_sizes[2];                // N_OUT = 20,000
    const int n_batch = in_sizes[0] / n_elem;       // B     = 64

    int* seg_start = (int*)d_ws;                    // (n_seg + 1) ints = ~80 KB

    {   // Pass 1: segment boundaries (recomputed every call -> no cached state)
        const int threads = 256;
        const int blocks  = (n_seg + 1 + threads - 1) / threads;
        seg_bounds_kernel<<<blocks, threads, 0, stream>>>(mask_col, seg_start,
                                                          n_elem, n_seg);
    }
    {   // Pass 2: one thread per (b, s)
        const int total   = n_batch * n_seg;        // 1,280,000
        const int threads = 256;
        const int blocks  = (total + threads - 1) / threads;
        seg_dot_kernel<<<blocks, threads, 0, stream>>>(
            x, kern, bias, mask_row, seg_start, out, n_elem, n_seg, n_batch);
    }
}